// EmbeddingBagCollection_16320875724852
// MI455X (gfx1250) — compile-verified
//
#include <hip/hip_runtime.h>
#include <hip/hip_bf16.h>

typedef __attribute__((ext_vector_type(2))) float v2f;
typedef __attribute__((ext_vector_type(8))) float v8f;

#define NBAGS 8192
#define DIM   64

// One wave32 per bag. Pooling done as D = ones(16x4) x B(4x16) + C with
// V_WMMA_F32_16X16X4_F32; every row of D holds the column sum, so after the
// loop any accumulator VGPR0 element holds the bag-sum for dim (chunk*16 + lane&15).
// base/cnt are forced into SGPRs so the K-loop is scalar-controlled and EXEC
// is provably all-ones at every WMMA (ISA requirement).
__global__ __launch_bounds__(256) void ebc_pool_wmma(
    const int* __restrict__ ids0, const int* __restrict__ off0, const float* __restrict__ tab0, int tot0,
    const int* __restrict__ ids1, const int* __restrict__ off1, const float* __restrict__ tab1, int tot1,
    const int* __restrict__ ids2, const int* __restrict__ off2, const float* __restrict__ tab2, int tot2,
    const int* __restrict__ ids3, const int* __restrict__ off3, const float* __restrict__ tab3, int tot3,
    float* __restrict__ out)
{
    const int tid  = blockIdx.x * blockDim.x + threadIdx.x;
    const int wave = tid >> 5;           // grid sized exactly: 0 .. 4*NBAGS-1
    const int lane = tid & 31;

    const int t   = wave >> 13;          // table index 0..3 (wave-uniform)
    const int bag = wave & (NBAGS - 1);

    const int* ids; const int* offs; const float* tab; int total;
    if (t == 0)      { ids = ids0; offs = off0; tab = tab0; total = tot0; }
    else if (t == 1) { ids = ids1; offs = off1; tab = tab1; total = tot1; }
    else if (t == 2) { ids = ids2; offs = off2; tab = tab2; total = tot2; }
    else             { ids = ids3; offs = off3; tab = tab3; total = tot3; }

    // Wave-uniform by construction; readfirstlane makes them SGPRs so the
    // pooling loop is scalar-controlled (no EXEC masking around the WMMAs).
    const int base = __builtin_amdgcn_readfirstlane(offs[bag]);
    const int cnt  = __builtin_amdgcn_readfirstlane(offs[bag + 1]) - base;

    // B-matrix lane roles: lanes 0-15 carry rows l+0 (b.x) / l+1 (b.y),
    // lanes 16-31 carry rows l+2 / l+3; column n = lane & 15. The ones-trick
    // is invariant to the exact K <-> (VGPR, lane-half) permutation.
    const int half = lane >> 4;
    const int col  = lane & 15;
    const int r0   = 2 * half;
    const int r1   = 2 * half + 1;

    v2f ones; ones.x = 1.0f; ones.y = 1.0f;   // A = all-ones 16x4
    v8f acc0 = {}; v8f acc1 = {}; v8f acc2 = {}; v8f acc3 = {};

    for (int l = 0; l < cnt; l += 4) {
        const int row0 = l + r0;
        const int row1 = l + r1;
        const float m0 = (row0 < cnt) ? 1.0f : 0.0f;   // remainder masking, branchless
        const float m1 = (row1 < cnt) ? 1.0f : 0.0f;

        int s0 = base + row0; s0 = (s0 < total) ? s0 : (total - 1);  // clamp: never OOB
        int s1 = base + row1; s1 = (s1 < total) ? s1 : (total - 1);
        const long e0 = (long)ids[s0] * DIM;
        const long e1 = (long)ids[s1] * DIM;

        // Prefetch next iteration's rows (ids are contiguous -> addresses known
        // one step ahead). 16 lanes at 16B stride cover the whole 256B row.
        int p0 = s0 + 4; p0 = (p0 < total) ? p0 : (total - 1);
        int p1 = s1 + 4; p1 = (p1 < total) ? p1 : (total - 1);
        __builtin_prefetch(tab + (long)ids[p0] * DIM + col * 4, 0, 3);
        __builtin_prefetch(tab + (long)ids[p1] * DIM + col * 4, 0, 3);

        v2f b;
        // dim chunk 0: cols 0..15
        b.x = tab[e0 +  0 + col] * m0;
        b.y = tab[e1 +  0 + col] * m1;
        acc0 = __builtin_amdgcn_wmma_f32_16x16x4_f32(false, ones, false, b,
                                                     (short)0, acc0, false, false);
        // dim chunk 1: cols 16..31
        b.x = tab[e0 + 16 + col] * m0;
        b.y = tab[e1 + 16 + col] * m1;
        acc1 = __builtin_amdgcn_wmma_f32_16x16x4_f32(false, ones, false, b,
                                                     (short)0, acc1, false, false);
        // dim chunk 2: cols 32..47
        b.x = tab[e0 + 32 + col] * m0;
        b.y = tab[e1 + 32 + col] * m1;
        acc2 = __builtin_amdgcn_wmma_f32_16x16x4_f32(false, ones, false, b,
                                                     (short)0, acc2, false, false);
        // dim chunk 3: cols 48..63
        b.x = tab[e0 + 48 + col] * m0;
        b.y = tab[e1 + 48 + col] * m1;
        acc3 = __builtin_amdgcn_wmma_f32_16x16x4_f32(false, ones, false, b,
                                                     (short)0, acc3, false, false);
    }

    // mean (empty bag -> 0 via max(cnt,1), matching torch semantics)
    const float inv = 1.0f / fmaxf((float)cnt, 1.0f);

    // Every lane's acc*[0] holds the sum for dim (chunk*16 + col) — rows of D
    // are identical. Split stores across the two lane halves.
    const size_t ob = ((size_t)t * NBAGS + (size_t)bag) * DIM + (size_t)col;
    if (half == 0) {
        out[ob +  0] = acc0[0] * inv;
        out[ob + 16] = acc1[0] * inv;
    } else {
        out[ob + 32] = acc2[0] * inv;
        out[ob + 48] = acc3[0] * inv;
    }
}

extern "C" void kernel_launch(void* const* d_in, const int* in_sizes, int n_in,
                              void* d_out, int out_size, void* d_ws, size_t ws_size,
                              hipStream_t stream) {
    (void)n_in; (void)out_size; (void)d_ws; (void)ws_size;
    // setup_inputs order: per table (uid, iid, cate, tag): ids, offsets, table
    const int*   ids0 = (const int*)  d_in[0];
    const int*   off0 = (const int*)  d_in[1];
    const float* tab0 = (const float*)d_in[2];
    const int*   ids1 = (const int*)  d_in[3];
    const int*   off1 = (const int*)  d_in[4];
    const float* tab1 = (const float*)d_in[5];
    const int*   ids2 = (const int*)  d_in[6];
    const int*   off2 = (const int*)  d_in[7];
    const float* tab2 = (const float*)d_in[8];
    const int*   ids3 = (const int*)  d_in[9];
    const int*   off3 = (const int*)  d_in[10];
    const float* tab3 = (const float*)d_in[11];

    const int tot0 = in_sizes[0];
    const int tot1 = in_sizes[3];
    const int tot2 = in_sizes[6];
    const int tot3 = in_sizes[9];

    // 4 tables * 8192 bags = 32768 waves; 8 waves (256 threads) per block.
    dim3 grid(4 * NBAGS / 8);
    dim3 block(256);
    hipLaunchKernelGGL(ebc_pool_wmma, grid, block, 0, stream,
                       ids0, off0, tab0, tot0,
                       ids1, off1, tab1, tot1,
                       ids2, off2, tab2, tot2,
                       ids3, off3, tab3, tot3,
                       (float*)d_out);
}